// EdgewiseGrad_84086869721639
// MI455X (gfx1250) — compile-verified
//
#include <hip/hip_runtime.h>
#include <hip/hip_bf16.h>
#include <cstdint>

// ---------------------------------------------------------------------------
// EdgewiseGrad on MI455X (gfx1250)
//   forces[N,3]  = scatter(+fij by idx0) + scatter(-fij by idx1)
//   stress[G,6]  = (sum over edges of virial, grouped by batch[idx1]) / |det(cell)|
// Bandwidth-bound: ~205MB streamed -> ~9us floor at 23.3 TB/s.
// gfx1250 async global->LDS copies (ASYNCcnt) double-buffer 128-edge tiles,
// marked non-temporal so the one-shot stream doesn't evict the L2-resident
// force array / batch table that the atomics keep hitting.
// Virial accumulates in a per-block 512x6 LDS table (ds_add_f32), flushed once
// per block; 8 waves share one table to halve flush traffic vs 4-wave blocks.
// ---------------------------------------------------------------------------

#define WAVES_PER_BLOCK 8
#define BLOCK_THREADS   (WAVES_PER_BLOCK * 32)
#define MAX_BLOCKS      1280
#define TILE_EDGES      128
#define TILE_F_BYTES    (TILE_EDGES * 12)   // 1536 B (float3 per edge)
#define TILE_I_BYTES    (TILE_EDGES * 4)    // 512 B  (int per edge)
#define BUF_BYTES       (2 * TILE_F_BYTES + 2 * TILE_I_BYTES)  // 4096 B
#define FIJ_OFF 0
#define EV_OFF  (TILE_F_BYTES)
#define I0_OFF  (2 * TILE_F_BYTES)
#define I1_OFF  (2 * TILE_F_BYTES + TILE_I_BYTES)
#define MAX_G_LDS 512

typedef __attribute__((address_space(3))) const void lds_cvoid_t;

__device__ __forceinline__ unsigned lds_addr_of(const void* p) {
    // generic -> LDS address space; ptrtoint gives workgroup-relative LDS byte addr
    return (unsigned)(uintptr_t)(lds_cvoid_t*)p;
}

__device__ __forceinline__ void async_b128(unsigned lds, const void* gaddr) {
    // each active lane copies 16B global->LDS; tracked by ASYNCcnt.
    // NT hint: stream is read-once, keep L2 for the atomic working set.
    asm volatile("global_load_async_to_lds_b128 %0, %1, off th:TH_LOAD_NT"
                 :: "v"(lds), "v"((uint64_t)(uintptr_t)gaddr)
                 : "memory");
}

template <int N>
__device__ __forceinline__ void wait_asynccnt() {
#if __has_builtin(__builtin_amdgcn_s_wait_asynccnt)
    __builtin_amdgcn_s_wait_asynccnt(N);
#else
    asm volatile("s_wait_asynccnt %0" :: "i"(N) : "memory");
#endif
}

__device__ __forceinline__ void atomAddGlobalF(float* p, float v) {
    // hardware global_atomic_add_f32, not a CAS loop
    unsafeAtomicAdd(p, v);
}

__global__ __launch_bounds__(BLOCK_THREADS)
void edgewise_scatter_kernel(const float* __restrict__ fij,
                             const float* __restrict__ ev,
                             const int*   __restrict__ idx0,
                             const int*   __restrict__ idx1,
                             const int*   __restrict__ batch,
                             float*       __restrict__ forces,
                             float*       __restrict__ gacc,   // [G*6] global accumulator
                             int fullTiles, int nGraphs)
{
    __shared__ __align__(16) char  stage[WAVES_PER_BLOCK * 2 * BUF_BYTES]; // 64 KB
    __shared__ float sacc[MAX_G_LDS * 6];                                  // 12 KB

    const int tid = threadIdx.x;
    for (int i = tid; i < MAX_G_LDS * 6; i += BLOCK_THREADS) sacc[i] = 0.0f;
    __syncthreads();

    const int lane = tid & 31;
    const int wave = tid >> 5;
    const int waveGlobal = blockIdx.x * WAVES_PER_BLOCK + wave;
    const int totalWaves = gridDim.x * WAVES_PER_BLOCK;

    const char* bufP[2] = { stage + (wave * 2 + 0) * BUF_BYTES,
                            stage + (wave * 2 + 1) * BUF_BYTES };
    const unsigned bufL[2] = { lds_addr_of(bufP[0]), lds_addr_of(bufP[1]) };

    const char* fij_c = (const char*)fij;
    const char* ev_c  = (const char*)ev;
    const char* i0_c  = (const char*)idx0;
    const char* i1_c  = (const char*)idx1;

    auto issue = [&](int tile, unsigned lbase) {
        const unsigned lo = (unsigned)(lane * 16);
        const char* gf = fij_c + (size_t)tile * TILE_F_BYTES;
        const char* ge = ev_c  + (size_t)tile * TILE_F_BYTES;
        async_b128(lbase + FIJ_OFF + 0 * 512 + lo, gf + 0 * 512 + lo);
        async_b128(lbase + FIJ_OFF + 1 * 512 + lo, gf + 1 * 512 + lo);
        async_b128(lbase + FIJ_OFF + 2 * 512 + lo, gf + 2 * 512 + lo);
        async_b128(lbase + EV_OFF  + 0 * 512 + lo, ge + 0 * 512 + lo);
        async_b128(lbase + EV_OFF  + 1 * 512 + lo, ge + 1 * 512 + lo);
        async_b128(lbase + EV_OFF  + 2 * 512 + lo, ge + 2 * 512 + lo);
        async_b128(lbase + I0_OFF + lo, i0_c + (size_t)tile * TILE_I_BYTES + lo);
        async_b128(lbase + I1_OFF + lo, i1_c + (size_t)tile * TILE_I_BYTES + lo);
    };

    auto process = [&](const char* sbuf) {
        const float* lf  = (const float*)(sbuf + FIJ_OFF);
        const float* le  = (const float*)(sbuf + EV_OFF);
        const int*   li0 = (const int*)(sbuf + I0_OFF);
        const int*   li1 = (const int*)(sbuf + I1_OFF);
#pragma unroll
        for (int k = 0; k < TILE_EDGES / 32; ++k) {
            const int j = lane + k * 32;
            const float fx = lf[j * 3 + 0], fy = lf[j * 3 + 1], fz = lf[j * 3 + 2];
            const float ex = le[j * 3 + 0], ey = le[j * 3 + 1], ez = le[j * 3 + 2];
            const int i0 = li0[j];
            const int i1 = li1[j];
            atomAddGlobalF(forces + (size_t)i0 * 3 + 0,  fx);
            atomAddGlobalF(forces + (size_t)i0 * 3 + 1,  fy);
            atomAddGlobalF(forces + (size_t)i0 * 3 + 2,  fz);
            atomAddGlobalF(forces + (size_t)i1 * 3 + 0, -fx);
            atomAddGlobalF(forces + (size_t)i1 * 3 + 1, -fy);
            atomAddGlobalF(forces + (size_t)i1 * 3 + 2, -fz);
            const int g = batch[i1];
            if (g < MAX_G_LDS) {
                float* s = sacc + g * 6;
                atomicAdd(s + 0, ex * fx);   // xx
                atomicAdd(s + 1, ey * fy);   // yy
                atomicAdd(s + 2, ez * fz);   // zz
                atomicAdd(s + 3, ey * fz);   // yz (s23)
                atomicAdd(s + 4, ez * fx);   // zx (s31)
                atomicAdd(s + 5, ex * fy);   // xy (s12)
            } else {
                float* s = gacc + (size_t)g * 6;
                atomAddGlobalF(s + 0, ex * fx);
                atomAddGlobalF(s + 1, ey * fy);
                atomAddGlobalF(s + 2, ez * fz);
                atomAddGlobalF(s + 3, ey * fz);
                atomAddGlobalF(s + 4, ez * fx);
                atomAddGlobalF(s + 5, ex * fy);
            }
        }
    };

    // double-buffered async pipeline over this wave's tiles
    int t = waveGlobal;
    if (t < fullTiles) issue(t, bufL[0]);
    int parity = 0;
    for (; t < fullTiles; t += totalWaves, parity ^= 1) {
        const int tn = t + totalWaves;
        if (tn < fullTiles) {
            issue(tn, bufL[parity ^ 1]);
            wait_asynccnt<8>();   // in-order completion: current tile's 8 copies done
        } else {
            wait_asynccnt<0>();
        }
        process(bufP[parity]);
    }

    __syncthreads();
    const int flushN = (nGraphs < MAX_G_LDS ? nGraphs : MAX_G_LDS) * 6;
    for (int i = tid; i < flushN; i += BLOCK_THREADS)
        atomAddGlobalF(gacc + i, sacc[i]);
}

__global__ void edgewise_tail_kernel(const float* __restrict__ fij,
                                     const float* __restrict__ ev,
                                     const int*   __restrict__ idx0,
                                     const int*   __restrict__ idx1,
                                     const int*   __restrict__ batch,
                                     float*       __restrict__ forces,
                                     float*       __restrict__ gacc,
                                     int start, int E)
{
    const int e = start + blockIdx.x * blockDim.x + threadIdx.x;
    if (e >= E) return;
    const float fx = fij[(size_t)e * 3 + 0], fy = fij[(size_t)e * 3 + 1], fz = fij[(size_t)e * 3 + 2];
    const float ex = ev[(size_t)e * 3 + 0],  ey = ev[(size_t)e * 3 + 1],  ez = ev[(size_t)e * 3 + 2];
    const int i0 = idx0[e];
    const int i1 = idx1[e];
    atomAddGlobalF(forces + (size_t)i0 * 3 + 0,  fx);
    atomAddGlobalF(forces + (size_t)i0 * 3 + 1,  fy);
    atomAddGlobalF(forces + (size_t)i0 * 3 + 2,  fz);
    atomAddGlobalF(forces + (size_t)i1 * 3 + 0, -fx);
    atomAddGlobalF(forces + (size_t)i1 * 3 + 1, -fy);
    atomAddGlobalF(forces + (size_t)i1 * 3 + 2, -fz);
    const int g = batch[i1];
    float* s = gacc + (size_t)g * 6;
    atomAddGlobalF(s + 0, ex * fx);
    atomAddGlobalF(s + 1, ey * fy);
    atomAddGlobalF(s + 2, ez * fz);
    atomAddGlobalF(s + 3, ey * fz);
    atomAddGlobalF(s + 4, ez * fx);
    atomAddGlobalF(s + 5, ex * fy);
}

__global__ void edgewise_zero_kernel(float* forces, int nF, float* gacc, int nA)
{
    const int i = blockIdx.x * blockDim.x + threadIdx.x;
    if (i < nF) forces[i] = 0.0f;
    if (i < nA) gacc[i] = 0.0f;
}

__global__ void edgewise_finalize_kernel(const float* __restrict__ cell,
                                         const float* __restrict__ gacc,
                                         float* __restrict__ stress, int G)
{
    const int g = blockIdx.x * blockDim.x + threadIdx.x;
    if (g >= G) return;
    const float* c = cell + (size_t)g * 9;
    const float det = c[0] * (c[4] * c[8] - c[5] * c[7])
                    - c[1] * (c[3] * c[8] - c[5] * c[6])
                    + c[2] * (c[3] * c[7] - c[4] * c[6]);
    const float inv = 1.0f / fabsf(det);
#pragma unroll
    for (int k = 0; k < 6; ++k)
        stress[(size_t)g * 6 + k] = gacc[(size_t)g * 6 + k] * inv;
}

extern "C" void kernel_launch(void* const* d_in, const int* in_sizes, int n_in,
                              void* d_out, int out_size, void* d_ws, size_t ws_size,
                              hipStream_t stream)
{
    (void)n_in; (void)out_size; (void)ws_size;
    const float* fij   = (const float*)d_in[0];
    const float* ev    = (const float*)d_in[1];
    const int*   eidx  = (const int*)d_in[2];
    const int*   batch = (const int*)d_in[3];
    const float* cell  = (const float*)d_in[4];

    const int E = in_sizes[0] / 3;
    const int N = in_sizes[3];
    const int G = in_sizes[4] / 9;

    float* forces = (float*)d_out;                    // [N*3]
    float* stress = (float*)d_out + (size_t)N * 3;    // [G*6]
    float* gacc   = (float*)d_ws;                     // [G*6] accumulator

    const int* idx0 = eidx;       // edge_index[0]
    const int* idx1 = eidx + E;   // edge_index[1]

    const int nF = N * 3;
    const int nA = G * 6;
    const int zt = nF > nA ? nF : nA;
    edgewise_zero_kernel<<<(zt + 255) / 256, 256, 0, stream>>>(forces, nF, gacc, nA);

    const int fullTiles = E / TILE_EDGES;
    int blocks = (fullTiles + WAVES_PER_BLOCK - 1) / WAVES_PER_BLOCK;
    if (blocks > MAX_BLOCKS) blocks = MAX_BLOCKS;
    if (blocks > 0) {
        edgewise_scatter_kernel<<<blocks, BLOCK_THREADS, 0, stream>>>(
            fij, ev, idx0, idx1, batch, forces, gacc, fullTiles, G);
    }

    const int rem = E - fullTiles * TILE_EDGES;
    if (rem > 0) {
        edgewise_tail_kernel<<<(rem + 127) / 128, 128, 0, stream>>>(
            fij, ev, idx0, idx1, batch, forces, gacc, fullTiles * TILE_EDGES, E);
    }

    edgewise_finalize_kernel<<<(G + 255) / 256, 256, 0, stream>>>(cell, gacc, stress, G);
}